// CTCLoss_16355235463906
// MI455X (gfx1250) — compile-verified
//
#include <hip/hip_runtime.h>
#include <hip/hip_bf16.h>
#include <stdint.h>

// Problem constants (from reference): T=2048, B=64, V=128, L=256
#define T_DIM 2048
#define B_DIM 64
#define V_DIM 128
#define L_DIM 256
#define S_DIM 513   // 2L+1
#define NEGF (-1.0e30f)

typedef float v2f __attribute__((ext_vector_type(2)));
typedef float v8f __attribute__((ext_vector_type(8)));

// Address-space-qualified pointer types for the async Global->LDS builtin.
typedef __attribute__((address_space(1))) int gas_int;  // global (AS1)
typedef __attribute__((address_space(3))) int las_int;  // LDS (AS3)

#if defined(__has_builtin)
#  if __has_builtin(__builtin_amdgcn_wmma_f32_16x16x4_f32)
#    define HAVE_WMMA_F32X4 1
#  endif
#  if __has_builtin(__builtin_amdgcn_global_load_async_to_lds_b32)
#    define HAVE_ASYNC_LDS 1
#  endif
#endif
#ifndef HAVE_WMMA_F32X4
#define HAVE_WMMA_F32X4 0
#endif
#ifndef HAVE_ASYNC_LDS
#define HAVE_ASYNC_LDS 0
#endif

// CDNA5 split dependency counters: async Global->LDS copies use ASYNCcnt.
#define S_WAIT_ASYNC0() asm volatile("s_wait_asynccnt 0x0" ::: "memory")
#define S_WAIT_ASYNC1() asm volatile("s_wait_asynccnt 0x1" ::: "memory")

__device__ __forceinline__ float lae(float x, float y) {
  // logaddexp with -1e30 acting as -inf
  const float m = fmaxf(x, y);
  const float d = fminf(x, y) - m;
  const float r = m + log1pf(__expf(d));
  return (m <= -1.0e29f) ? NEGF : r;
}

// ---------------------------------------------------------------------------
// Kernel 1: per-(t,b) row log-sum-exp of acts[t,b,:] (V=128), 16 rows per wave.
// Row sums of exp() are computed with chained V_WMMA_F32_16X16X4_F32 against an
// all-ones B matrix: D[m,n] = sum_k A[m,k] -> exact f32 row sums on the matrix
// pipe. A-layout (16x4 f32): lanes 0-15 hold K={0,1}, lanes 16-31 hold K={2,3}.
// ---------------------------------------------------------------------------
__global__ __launch_bounds__(256) void ctc_lse_kernel(
    const float* __restrict__ acts, float* __restrict__ lse,
    float* __restrict__ out) {
  if (blockIdx.x == 0 && threadIdx.x == 0) out[0] = 0.0f;  // accumulator init

  const int wid  = threadIdx.x >> 5;
  const int lane = threadIdx.x & 31;
  const int r    = lane & 15;   // row within the wave's 16-row tile
  const int hi   = lane >> 4;   // which K-half of each 4-wide chunk
  const int rowBase = blockIdx.x * 128 + wid * 16;  // T*B = 131072 = 1024*128 exact
  const float* rowp = acts + (size_t)(rowBase + r) * V_DIM + 2 * hi;

  // Pass 1: row max (stripes combined across the two half-waves).
  float pm = -3.0e38f;
#pragma unroll
  for (int k = 0; k < 32; ++k) {
    const float2 x = *(const float2*)(rowp + 4 * k);
    pm = fmaxf(pm, fmaxf(x.x, x.y));
  }
  pm = fmaxf(pm, __shfl_xor(pm, 16));

  float rows[8];
#if HAVE_WMMA_F32X4
  // Pass 2: WMMA row-sum of exp(x - max), 32 chained K=4 steps (K total = 128).
  v8f acc = {0.f, 0.f, 0.f, 0.f, 0.f, 0.f, 0.f, 0.f};
  const v2f ones = {1.0f, 1.0f};
#pragma unroll
  for (int k = 0; k < 32; ++k) {
    const float2 x = *(const float2*)(rowp + 4 * k);
    v2f a;
    a.x = __expf(x.x - pm);
    a.y = __expf(x.y - pm);
    acc = __builtin_amdgcn_wmma_f32_16x16x4_f32(
        /*neg_a=*/false, a, /*neg_b=*/false, ones,
        /*c_mod=*/(short)0, acc, /*reuse_a=*/false, /*reuse_b=*/false);
  }
  // D layout: VGPR i holds rowsum(i + 8*hi), replicated across N (lanes).
#pragma unroll
  for (int i = 0; i < 8; ++i) rows[i] = acc[i];
#else
  float part = 0.f;
#pragma unroll
  for (int k = 0; k < 32; ++k) {
    const float2 x = *(const float2*)(rowp + 4 * k);
    part += __expf(x.x - pm) + __expf(x.y - pm);
  }
  const float tot = part + __shfl_xor(part, 16);
#pragma unroll
  for (int i = 0; i < 8; ++i) rows[i] = __shfl(tot, i + 8 * hi);
#endif

  // Lanes with N==0 (lanes 0 and 16) publish rows {0..7} / {8..15}.
#pragma unroll
  for (int i = 0; i < 8; ++i) {
    const float mv = __shfl(pm, i + 8 * hi);  // max of row i+8*hi lives in lane i+8*hi
    if (r == 0) lse[rowBase + i + 8 * hi] = mv + __logf(rows[i]);
  }
}

// ---------------------------------------------------------------------------
// Kernel 2: CTC forward-alpha scan. One block per batch (64 blocks),
// 544 threads = 17 wave32 waves, thread s owns state s (s < 513).
// acts row for t+1 is double-buffered into LDS with async Global->LDS copies
// (ASYNCcnt), overlapped with the alpha update for t.
// ---------------------------------------------------------------------------
__global__ __launch_bounds__(544) void ctc_alpha_kernel(
    const float* __restrict__ acts, const float* __restrict__ lse,
    const int* __restrict__ labels, const int* __restrict__ act_lens,
    const int* __restrict__ label_lens, float* __restrict__ out) {
  const int b   = blockIdx.x;
  const int tid = threadIdx.x;

  __shared__ float alphaBuf[2 + S_DIM];  // [0..1] = permanent -inf pad
  __shared__ float actsBuf[2][V_DIM];    // double-buffered acts[t,b,:]
  __shared__ float lseBuf[2];

  const int act_len = act_lens[b];
  const int lab_len = label_lens[b];

  // Per-state constants: extended label and skip-allowed flag.
  int ext = 0;          // BLANK = 0 for even states
  bool allow = false;   // skip transition s-2 -> s
  if (tid < S_DIM && (tid & 1)) {
    ext = labels[b * L_DIM + (tid >> 1)];
    if (tid >= 3) allow = (ext != labels[b * L_DIM + ((tid - 2) >> 1)]);
  }

  auto load_acts = [&](int t, int buf) {
    if (tid < V_DIM) {
      const size_t off = ((size_t)t * B_DIM + b) * V_DIM + tid;
#if HAVE_ASYNC_LDS
      __builtin_amdgcn_global_load_async_to_lds_b32(
          (gas_int*)(acts + off),
          (las_int*)&actsBuf[buf][tid],
          /*offset=*/0, /*cpol=*/0);
#else
      actsBuf[buf][tid] = acts[off];
#endif
    }
    if (tid == 0) lseBuf[buf] = lse[t * B_DIM + b];
  };

  // t = 0: stage acts row, init alpha.
  load_acts(0, 0);
  S_WAIT_ASYNC0();
  __syncthreads();

  if (tid < 2) alphaBuf[tid] = NEGF;
  if (tid < S_DIM) {
    float v = NEGF;
    if (tid == 0)      v = actsBuf[0][0]   - lseBuf[0];
    else if (tid == 1) v = actsBuf[0][ext] - lseBuf[0];
    alphaBuf[2 + tid] = v;
  }
  load_acts(1, 1);      // prefetch t=1 into buf1 (fresh buffer, no readers yet)
  __syncthreads();      // init reads of buf0 complete before iter-1 overwrites it

  for (int t = 1; t < T_DIM; ++t) {
    const int cur = t & 1;
    if (t + 1 < T_DIM) {
      load_acts(t + 1, cur ^ 1);  // safe: previous iter's readers barriered out
      S_WAIT_ASYNC1();            // t's copy (older, in-order) has landed
    } else {
      S_WAIT_ASYNC0();
    }
    __syncthreads();  // B1: actsBuf[cur]/lseBuf[cur] + alpha(t-1) visible

    float newv = NEGF;
    if (tid < S_DIM) {
      const float a0 = alphaBuf[2 + tid];
      float c = lae(a0, alphaBuf[1 + tid]);
      if (allow) c = lae(c, alphaBuf[tid]);
      const float lp = actsBuf[cur][ext] - lseBuf[cur];
      newv = (t < act_len) ? (c + lp) : a0;
    }
    __syncthreads();  // B2: all alpha reads done before in-place update
    if (tid < S_DIM) alphaBuf[2 + tid] = newv;
  }
  __syncthreads();

  if (tid == 0) {
    const float l1 = alphaBuf[2 + 2 * lab_len];
    const float l2 = alphaBuf[2 + 2 * lab_len - 1];
    atomicAdd(out, -lae(l1, l2));
  }
}

// ---------------------------------------------------------------------------
extern "C" void kernel_launch(void* const* d_in, const int* in_sizes, int n_in,
                              void* d_out, int out_size, void* d_ws,
                              size_t ws_size, hipStream_t stream) {
  (void)in_sizes; (void)n_in; (void)out_size; (void)ws_size;
  const float* acts      = (const float*)d_in[0];
  const int*   labels    = (const int*)d_in[1];
  const int*   act_lens  = (const int*)d_in[2];
  const int*   label_lens= (const int*)d_in[3];
  float* out = (float*)d_out;
  float* lse = (float*)d_ws;  // T*B floats = 512 KB scratch

  // 131072 rows, 128 rows per 256-thread block (16 rows per wave).
  ctc_lse_kernel<<<dim3((T_DIM * B_DIM) / 128), dim3(256), 0, stream>>>(acts, lse, out);
  // One block per batch; 17 wave32 waves cover S=513 states.
  ctc_alpha_kernel<<<dim3(B_DIM), dim3(544), 0, stream>>>(acts, lse, labels,
                                                          act_lens, label_lens, out);
}